// BlastLinear_51857435132626
// MI455X (gfx1250) — compile-verified
//
#include <hip/hip_runtime.h>

typedef __attribute__((ext_vector_type(2))) float v2f;
typedef __attribute__((ext_vector_type(8))) float v8f;

#define IN_F 4096
#define OUT_F 4096
#define PBLK 4
#define QBLK 4
#define RNK 512
#define CIN (IN_F / QBLK)    // 1024
#define OBLK (OUT_F / PBLK)  // 1024

#define TM 64        // output tile rows per block
#define TN 64        // output tile cols per block
#define TK 32        // K slab staged in LDS
#define LDST 36      // LDS row stride in floats (conflict-free, 16B aligned)

// ---------------------------------------------------------------------------
// Stage 1: y[q, n, r] = sum_c x[n, q*1024 + c] * C[q, r, c]
// grid: (N/64, R/64, Q), block: 256 (8 waves)
// ---------------------------------------------------------------------------
__global__ __launch_bounds__(256) void k_stage1(const float* __restrict__ x,
                                                const float* __restrict__ C,
                                                float* __restrict__ y,
                                                int N) {
  __shared__ __align__(16) float xs[TM * LDST];
  __shared__ __align__(16) float cs[TN * LDST];

  const int n0 = blockIdx.x * TM;
  const int r0 = blockIdx.y * TN;
  const int q  = blockIdx.z;

  const int tid  = threadIdx.x;
  const int lane = tid & 31;
  const int wave = tid >> 5;
  const int wm   = wave >> 1;    // 0..3 : 16-row slice
  const int wn   = wave & 1;     // 0..1 : 32-col slice
  const int mi   = lane & 15;
  const int half = lane >> 4;

  v8f acc0 = {};
  v8f acc1 = {};

  const int lrow = tid >> 3;          // 0..31
  const int lcol = (tid & 7) * 4;     // 0,4,...,28

  const float* xg = x + (size_t)(n0 + lrow) * IN_F + (size_t)q * CIN + lcol;
  const float* cg = C + (size_t)q * RNK * CIN + (size_t)(r0 + lrow) * CIN + lcol;

  for (int ck = 0; ck < CIN; ck += TK) {
    // ---- stage x[64 x 32] and C[64 x 32] tiles ----
#pragma unroll
    for (int ph = 0; ph < 2; ++ph) {
      const float4 xv = *(const float4*)(xg + (size_t)(ph * 32) * IN_F + ck);
      const float4 cv = *(const float4*)(cg + (size_t)(ph * 32) * CIN + ck);
      *(float4*)&xs[(lrow + ph * 32) * LDST + lcol] = xv;
      *(float4*)&cs[(lrow + ph * 32) * LDST + lcol] = cv;
    }
    __syncthreads();

    // ---- 16x16x4 f32 WMMA over the slab ----
    const float* ap  = &xs[(wm * 16 + mi) * LDST];
    const float* bp0 = &cs[(wn * 32 + mi) * LDST];
    const float* bp1 = bp0 + 16 * LDST;
#pragma unroll
    for (int kk = 0; kk < TK; kk += 4) {
      const int c0 = kk + 2 * half;
      v2f a  = {ap[c0],  ap[c0 + 1]};
      v2f b0 = {bp0[c0], bp0[c0 + 1]};
      v2f b1 = {bp1[c0], bp1[c0 + 1]};
      acc0 = __builtin_amdgcn_wmma_f32_16x16x4_f32(false, a, false, b0,
                                                   (short)0, acc0, false, false);
      acc1 = __builtin_amdgcn_wmma_f32_16x16x4_f32(false, a, false, b1,
                                                   (short)0, acc1, false, false);
    }
    __syncthreads();
  }

  // ---- write y[q][n][r] ----
  float* yq = y + (size_t)q * N * RNK;
  const int rc0 = r0 + wn * 32 + mi;
#pragma unroll
  for (int i = 0; i < 8; ++i) {
    const size_t row = (size_t)(n0 + wm * 16 + half * 8 + i);
    yq[row * RNK + rc0]      = acc0[i];
    yq[row * RNK + rc0 + 16] = acc1[i];
  }
}

// ---------------------------------------------------------------------------
// Stage 2: z[p,n,r] = sum_q D[p,q,r] * y[q,n,r]   (formed in the staging loop)
//          out[n, p*1024 + j] = sum_r z[p,n,r] * B[p,j,r] + bias[p*1024+j]
// grid: (N/64, OBLK/64, P), block: 256 (8 waves)
// ---------------------------------------------------------------------------
__global__ __launch_bounds__(256) void k_stage2(const float* __restrict__ y,
                                                const float* __restrict__ B,
                                                const float* __restrict__ D,
                                                const float* __restrict__ bias,
                                                float* __restrict__ out,
                                                int N) {
  __shared__ __align__(16) float zs[TM * LDST];
  __shared__ __align__(16) float bs[TN * LDST];

  const int n0 = blockIdx.x * TM;
  const int j0 = blockIdx.y * TN;
  const int p  = blockIdx.z;

  const int tid  = threadIdx.x;
  const int lane = tid & 31;
  const int wave = tid >> 5;
  const int wm   = wave >> 1;
  const int wn   = wave & 1;
  const int mi   = lane & 15;
  const int half = lane >> 4;

  v8f acc0 = {};
  v8f acc1 = {};

  const int lrow = tid >> 3;
  const int lcol = (tid & 7) * 4;

  for (int rk = 0; rk < RNK; rk += TK) {
    // ---- stage z[64 x 32] (fused D-contraction over q) and B[64 x 32] ----
#pragma unroll
    for (int ph = 0; ph < 2; ++ph) {
      const int row = lrow + ph * 32;
      float4 zv = {0.f, 0.f, 0.f, 0.f};
#pragma unroll
      for (int q = 0; q < QBLK; ++q) {
        const float4 yv = *(const float4*)(y + (size_t)q * N * RNK +
                                           (size_t)(n0 + row) * RNK + rk + lcol);
        const float4 dv = *(const float4*)(D + (size_t)(p * QBLK + q) * RNK +
                                           rk + lcol);
        zv.x += yv.x * dv.x;
        zv.y += yv.y * dv.y;
        zv.z += yv.z * dv.z;
        zv.w += yv.w * dv.w;
      }
      *(float4*)&zs[row * LDST + lcol] = zv;
      const float4 bv = *(const float4*)(B + ((size_t)p * OBLK + j0 + row) * RNK +
                                         rk + lcol);
      *(float4*)&bs[row * LDST + lcol] = bv;
    }
    __syncthreads();

    const float* ap  = &zs[(wm * 16 + mi) * LDST];
    const float* bp0 = &bs[(wn * 32 + mi) * LDST];
    const float* bp1 = bp0 + 16 * LDST;
#pragma unroll
    for (int kk = 0; kk < TK; kk += 4) {
      const int c0 = kk + 2 * half;
      v2f a  = {ap[c0],  ap[c0 + 1]};
      v2f b0 = {bp0[c0], bp0[c0 + 1]};
      v2f b1 = {bp1[c0], bp1[c0 + 1]};
      acc0 = __builtin_amdgcn_wmma_f32_16x16x4_f32(false, a, false, b0,
                                                   (short)0, acc0, false, false);
      acc1 = __builtin_amdgcn_wmma_f32_16x16x4_f32(false, a, false, b1,
                                                   (short)0, acc1, false, false);
    }
    __syncthreads();
  }

  // ---- epilogue: bias + interleaved store out[n, p*1024 + j] ----
  const int jc0 = j0 + wn * 32 + mi;
  const float bias0 = bias[p * OBLK + jc0];
  const float bias1 = bias[p * OBLK + jc0 + 16];
#pragma unroll
  for (int i = 0; i < 8; ++i) {
    const size_t row = (size_t)(n0 + wm * 16 + half * 8 + i);
    out[row * OUT_F + p * OBLK + jc0]      = acc0[i] + bias0;
    out[row * OUT_F + p * OBLK + jc0 + 16] = acc1[i] + bias1;
  }
}

// ---------------------------------------------------------------------------
extern "C" void kernel_launch(void* const* d_in, const int* in_sizes, int n_in,
                              void* d_out, int out_size, void* d_ws, size_t ws_size,
                              hipStream_t stream) {
  const float* x    = (const float*)d_in[0];
  const float* B    = (const float*)d_in[1];
  const float* C    = (const float*)d_in[2];
  const float* D    = (const float*)d_in[3];
  const float* bias = (const float*)d_in[4];
  float* out = (float*)d_out;
  float* y   = (float*)d_ws;  // Q*N*R fp32 = 64 MB scratch

  const int N = in_sizes[0] / IN_F;  // 8192

  dim3 g1(N / TM, RNK / TN, QBLK);   // (128, 8, 4)
  dim3 g2(N / TM, OBLK / TN, PBLK);  // (128, 16, 4)

  k_stage1<<<g1, 256, 0, stream>>>(x, C, y, N);
  k_stage2<<<g2, 256, 0, stream>>>(y, B, D, bias, out, N);
}